// CorrGame_3109556322609
// MI455X (gfx1250) — compile-verified
//
#include <hip/hip_runtime.h>
#include <stdint.h>

// Problem constants (match reference)
#define N_DIM 128
#define K_DIM 32
#define T_DIM 1000000
#define TC    64                 // T-columns per LDS tile
#define NTILES (T_DIM / TC)      // 15625 (exact)
#define NWG   1024

// LDS pitches (TDM pad: +1 dword per row -> conflict-free column reads)
#define XPITCH 65
#define WPITCH 129
#define MPITCH 33

// LDS partition (float offsets)
#define XS0_OFF 0
#define XS1_OFF (XPITCH * N_DIM)
#define YS0_OFF (2 * XPITCH * N_DIM)
#define YS1_OFF (2 * XPITCH * N_DIM + XPITCH * K_DIM)
#define W_OFF   (2 * XPITCH * N_DIM + 2 * XPITCH * K_DIM)
#define M_OFF   (W_OFF + WPITCH * K_DIM)
#define SMEM_FLOATS (M_OFF + MPITCH * K_DIM)   // 25984 floats = 103936 B < 320 KB

typedef __attribute__((ext_vector_type(2))) float    v2f;
typedef __attribute__((ext_vector_type(8))) float    v8f;
typedef __attribute__((ext_vector_type(4))) uint32_t su4;
typedef __attribute__((ext_vector_type(8))) uint32_t su8;

__device__ __forceinline__ v8f wmma_f32(v2f a, v2f b, v8f c) {
  // D = A(16x4) * B(4x16) + C(16x16), fp32
  return __builtin_amdgcn_wmma_f32_16x16x4_f32(
      false, a, false, b, (short)0, c, false, false);
}

// 2D TDM load: global (row-major, stride0 elems between rows) -> LDS,
// pad_enable=1, pad_amount=1 dword, interval = 2^(pad_code+1) dwords.
__device__ __forceinline__ void tdm_load_2d(uint32_t lds_byte_off,
                                            const float* gptr,
                                            uint32_t td0, uint32_t td1,
                                            uint32_t tile0, uint32_t tile1,
                                            uint64_t stride0,
                                            uint32_t pad_code) {
  uint64_t ga = (uint64_t)(uintptr_t)gptr;
  su4 g0;
  g0[0] = 1u;                                            // count=1, user mode
  g0[1] = lds_byte_off;                                  // lds_addr
  g0[2] = (uint32_t)ga;                                  // global_addr[31:0]
  g0[3] = (uint32_t)((ga >> 32) & 0x1FFFFFFu)            // global_addr[56:32]
        | (2u << 30);                                    // type = 2 (image)
  su8 g1;
  g1[0] = (2u << 16)                                     // data_size = 4B
        | (1u << 20)                                     // pad_enable
        | (pad_code << 22);                              // pad_interval; amount field 0 => 1 dword
  g1[1] = (td0 & 0xFFFFu) << 16;                         // tensor_dim0[15:0]
  g1[2] = (td0 >> 16) | ((td1 & 0xFFFFu) << 16);         // dim0 hi | dim1 lo
  g1[3] = (td1 >> 16) | (tile0 << 16);                   // dim1 hi | tile_dim0
  g1[4] = tile1 & 0xFFFFu;                               // tile_dim1 (tile_dim2 = 0)
  g1[5] = (uint32_t)stride0;                             // tensor_dim0_stride lo
  g1[6] = (uint32_t)((stride0 >> 32) & 0xFFFFu);         // stride hi (dim1_stride unused)
  g1[7] = 0u;
  asm volatile("tensor_load_to_lds %0, %1" :: "s"(g0), "s"(g1) : "memory");
}

__global__ void __launch_bounds__(256) corr_zero_ws(float* p, int n) {
  int i = blockIdx.x * blockDim.x + threadIdx.x;
  if (i < n) p[i] = 0.0f;
}

__global__ void __launch_bounds__(256)
corr_main(const float* __restrict__ X, const float* __restrict__ Y,
          const float* __restrict__ W, const float* __restrict__ M,
          float* __restrict__ Yout, float* __restrict__ wsW,
          float* __restrict__ wsM) {
  extern __shared__ float smem[];
  const int lane = threadIdx.x & 31;
  const int wid  = threadIdx.x >> 5;     // 8 waves
  const int l16  = lane & 15;
  const int lh   = lane >> 4;            // 0 | 1 (lane half)

  const float* Wl = smem + W_OFF;
  const float* Ml = smem + M_OFF;
  const uint32_t lds_base = (uint32_t)(uintptr_t)smem;   // flat LDS addr low 32b = LDS offset

  const int m0    = (wid & 1) * 16;   // Y_new row tile
  const int tcol0 = (wid >> 1) * 16;  // Y_new col tile within the 64-wide slab
  const int nb    = wid * 16;         // X-row tile for Y@X^T

  const int tile0 = blockIdx.x;
  if (wid == 0) {
    tdm_load_2d(lds_base + W_OFF * 4, W, N_DIM, K_DIM, N_DIM, K_DIM, N_DIM, 6); // pitch 129
    tdm_load_2d(lds_base + M_OFF * 4, M, K_DIM, K_DIM, K_DIM, K_DIM, K_DIM, 4); // pitch 33
    if (tile0 < NTILES) {
      tdm_load_2d(lds_base + XS0_OFF * 4, X + (size_t)tile0 * TC, T_DIM, N_DIM, TC, N_DIM, T_DIM, 5);
      tdm_load_2d(lds_base + YS0_OFF * 4, Y + (size_t)tile0 * TC, T_DIM, K_DIM, TC, K_DIM, T_DIM, 5);
    }
    __builtin_amdgcn_s_wait_tensorcnt(2);  // W, M landed (X0/Y0 may still be in flight)
  }
  __syncthreads();

  // ---- hoist loop-invariant A-fragments of W and -M into registers ----
  v2f wA[N_DIM / 4];
  #pragma unroll
  for (int k = 0; k < N_DIM / 4; ++k) {
    wA[k].x = Wl[(m0 + l16) * WPITCH + 4 * k + 2 * lh];
    wA[k].y = Wl[(m0 + l16) * WPITCH + 4 * k + 2 * lh + 1];
  }
  v2f mA[K_DIM / 4];                      // pre-negated: acc += (-M) @ Y
  #pragma unroll
  for (int k = 0; k < K_DIM / 4; ++k) {
    mA[k].x = -Ml[(m0 + l16) * MPITCH + 4 * k + 2 * lh];
    mA[k].y = -Ml[(m0 + l16) * MPITCH + 4 * k + 2 * lh + 1];
  }

  // persistent accumulators (reduced over all tiles this WG owns)
  v8f accW0 = {};  // Y@X^T tile (k-rows 0..15,  n-cols wid*16..)
  v8f accW1 = {};  // Y@X^T tile (k-rows 16..31, n-cols wid*16..)
  v8f accM  = {};  // waves 0..3: Y@Y^T tile ((wid>>1)*16.., (wid&1)*16..)

  int buf = 0;
  for (int tile = tile0; tile < NTILES; tile += gridDim.x, buf ^= 1) {
    const int nextTile = tile + (int)gridDim.x;
    if (wid == 0) {
      if (nextTile < NTILES) {
        const uint32_t xoff = buf ? XS0_OFF : XS1_OFF;
        const uint32_t yoff = buf ? YS0_OFF : YS1_OFF;
        tdm_load_2d(lds_base + xoff * 4, X + (size_t)nextTile * TC, T_DIM, N_DIM, TC, N_DIM, T_DIM, 5);
        tdm_load_2d(lds_base + yoff * 4, Y + (size_t)nextTile * TC, T_DIM, K_DIM, TC, K_DIM, T_DIM, 5);
        __builtin_amdgcn_s_wait_tensorcnt(2);  // current tile landed; next 2 in flight
      } else {
        __builtin_amdgcn_s_wait_tensorcnt(0);
      }
    }
    __syncthreads();

    const float* Xb = smem + (buf ? XS1_OFF : XS0_OFF);
    const float* Yb = smem + (buf ? YS1_OFF : YS0_OFF);

    // ---- Y_new tile: acc = (W@X - M@Y)[m0..m0+15, tcol0..tcol0+15] ----
    v8f acc = {};
    #pragma unroll
    for (int k = 0; k < N_DIM / 4; ++k) {
      const int n4 = 4 * k;
      v2f b;
      b.x = Xb[(n4 + 2 * lh) * XPITCH + tcol0 + l16];    // B: X[4x16] (row-read)
      b.y = Xb[(n4 + 2 * lh + 1) * XPITCH + tcol0 + l16];
      acc = wmma_f32(wA[k], b, acc);
    }
    #pragma unroll
    for (int k = 0; k < K_DIM / 4; ++k) {
      const int k4 = 4 * k;
      v2f b;
      b.x = Yb[(k4 + 2 * lh) * XPITCH + tcol0 + l16];
      b.y = Yb[(k4 + 2 * lh + 1) * XPITCH + tcol0 + l16];
      acc = wmma_f32(mA[k], b, acc);
    }
    {
      const size_t tg = (size_t)tile * TC + tcol0 + l16;
      #pragma unroll
      for (int v = 0; v < 8; ++v) {
        const int ml = m0 + v + 8 * lh;                  // C/D layout: vgpr v -> row v / v+8
        const float yold = Yb[ml * XPITCH + tcol0 + l16];
        Yout[(size_t)ml * T_DIM + tg] = yold + 0.01f * acc[v];
      }
    }

    // ---- accumulate Y@X^T (and Y@Y^T on waves 0..3) over 64 columns ----
    #pragma unroll
    for (int tc = 0; tc < TC / 4; ++tc) {
      const int t4 = 4 * tc;
      v2f a0, a1, b;
      a0.x = Yb[l16 * XPITCH + t4 + 2 * lh];             // A: Y rows 0..15, K=t
      a0.y = Yb[l16 * XPITCH + t4 + 2 * lh + 1];
      a1.x = Yb[(16 + l16) * XPITCH + t4 + 2 * lh];      // A: Y rows 16..31
      a1.y = Yb[(16 + l16) * XPITCH + t4 + 2 * lh + 1];
      b.x  = Xb[(nb + l16) * XPITCH + t4 + 2 * lh];      // B: X^T[4 x 16] (col-read, pitch 65)
      b.y  = Xb[(nb + l16) * XPITCH + t4 + 2 * lh + 1];
      accW0 = wmma_f32(a0, b, accW0);
      accW1 = wmma_f32(a1, b, accW1);
      if (wid < 4) {                                     // wave-uniform branch, EXEC stays full
        v2f by;
        by.x = Yb[((wid & 1) * 16 + l16) * XPITCH + t4 + 2 * lh];
        by.y = Yb[((wid & 1) * 16 + l16) * XPITCH + t4 + 2 * lh + 1];
        accM = wmma_f32((wid >> 1) ? a1 : a0, by, accM);
      }
    }
    __syncthreads();
  }

  // ---- flush reduction partials ----
  #pragma unroll
  for (int v = 0; v < 8; ++v) {
    const int kr = v + 8 * lh;
    unsafeAtomicAdd(&wsW[kr * N_DIM + nb + l16], accW0[v]);
    unsafeAtomicAdd(&wsW[(16 + kr) * N_DIM + nb + l16], accW1[v]);
  }
  if (wid < 4) {
    #pragma unroll
    for (int v = 0; v < 8; ++v) {
      const int kr = (wid >> 1) * 16 + v + 8 * lh;
      unsafeAtomicAdd(&wsM[kr * K_DIM + (wid & 1) * 16 + l16], accM[v]);
    }
  }
}

__global__ void __launch_bounds__(256)
corr_finalize(const float* __restrict__ W, const float* __restrict__ M,
              const float* __restrict__ wsW, const float* __restrict__ wsM,
              float* __restrict__ outWM) {
  const int i = blockIdx.x * blockDim.x + threadIdx.x;
  const float invT = 1.0f / (float)T_DIM;
  if (i < K_DIM * N_DIM) {
    const float w = W[i];
    outWM[i] = w + 0.1f * (wsW[i] * invT - w);           // W_new
  }
  const int j = i - K_DIM * N_DIM;
  if (j >= 0 && j < K_DIM * K_DIM) {
    const float m = M[j];
    outWM[K_DIM * N_DIM + j] = m + 0.05f * 0.5f * (wsM[j] * invT - m);  // M_new
  }
}

extern "C" void kernel_launch(void* const* d_in, const int* in_sizes, int n_in,
                              void* d_out, int out_size, void* d_ws, size_t ws_size,
                              hipStream_t stream) {
  const float* X = (const float*)d_in[0];   // [128, 1e6]
  const float* Y = (const float*)d_in[1];   // [32, 1e6]
  const float* W = (const float*)d_in[2];   // [32, 128]
  const float* M = (const float*)d_in[3];   // [32, 32]
  float* out = (float*)d_out;               // Y_new | W_new | M_new
  float* wsW = (float*)d_ws;                // [32*128] partials
  float* wsM = wsW + K_DIM * N_DIM;         // [32*32] partials

  corr_zero_ws<<<20, 256, 0, stream>>>(wsW, K_DIM * N_DIM + K_DIM * K_DIM);
  corr_main<<<NWG, 256, SMEM_FLOATS * sizeof(float), stream>>>(
      X, Y, W, M, out, wsW, wsM);
  corr_finalize<<<20, 256, 0, stream>>>(W, M, wsW, wsM,
                                        out + (size_t)K_DIM * T_DIM);
}